// LearnOVSTFT2MELChannel_21775484191426
// MI455X (gfx1250) — compile-verified
//
#include <hip/hip_runtime.h>
#include <hip/hip_bf16.h>

// Shapes from the reference
#define NFREQ 257
#define NFILTERS 64
#define NCH 2
#define D1 128
#define Bsz 8
#define Tsz 1000
#define AMIN 1e-10f
#define TOP_DB 80.0f

// WMMA vector types (CDNA5, wave32): f32 16x16x4 -> A,B = 2 VGPRs, C/D = 8 VGPRs
typedef __attribute__((ext_vector_type(2))) float v2f;
typedef __attribute__((ext_vector_type(8))) float v8f;

// ---- ordered-float encoding so unsigned atomicMax == float max ----
__device__ __forceinline__ unsigned enc_f32(float f) {
  unsigned u = __float_as_uint(f);
  return (u & 0x80000000u) ? ~u : (u | 0x80000000u);
}
__device__ __forceinline__ float dec_f32(unsigned u) {
  unsigned s = (u & 0x80000000u) ? (u & 0x7FFFFFFFu) : ~u;
  return __uint_as_float(s);
}

// ---------------- Phase 1: global max of 10*log10(max(x,AMIN)) ----------------
__global__ void mel_max_kernel(const float* __restrict__ x, int n, unsigned* gmax) {
  float m = -3.402823e38f;
  for (int i = blockIdx.x * blockDim.x + threadIdx.x; i < n;
       i += gridDim.x * blockDim.x) {
    float v = 10.0f * __log10f(fmaxf(x[i], AMIN));
    m = fmaxf(m, v);
  }
  for (int off = 16; off > 0; off >>= 1) m = fmaxf(m, __shfl_xor(m, off, 32));
  if ((threadIdx.x & 31) == 0) atomicMax(gmax, enc_f32(m));
}

// ---------------- Phase 2: feat[bt][c][k], c in {ch0, ch1, log(ch0^2+1e-8)} ----------------
__global__ void mel_feat_kernel(const float* __restrict__ x,
                                const unsigned* __restrict__ gmax,
                                float* __restrict__ feat, int n) {
  int i = blockIdx.x * blockDim.x + threadIdx.x;
  if (i >= n) return;
  int k = i % NFREQ;
  int bt = i / NFREQ;
  int b = bt / Tsz;
  int t = bt - b * Tsz;
  float floorv = dec_f32(*gmax) - TOP_DB;
  const float* xb = x + ((size_t)(b * NCH) * Tsz + t) * NFREQ + k;
  float v0 = fmaxf(10.0f * __log10f(fmaxf(xb[0], AMIN)), floorv);
  float v1 = fmaxf(10.0f * __log10f(fmaxf(xb[(size_t)Tsz * NFREQ], AMIN)), floorv);
  float v2 = __logf(v0 * v0 + 1e-8f);
  float* fp = feat + ((size_t)bt * 3) * NFREQ + k;
  fp[0]         = v0;
  fp[NFREQ]     = v1;
  fp[2 * NFREQ] = v2;
}

// ---------------- Phase 3: per-filter GEMM via V_WMMA_F32_16X16X4_F32 ----------------
// Block = 128 threads (4 waves). Block tile: 16 (b,t)-rows x 64 d-columns, all 64 filters.
// Two filters are processed per staging round -> each wave runs 2 independent WMMA
// accumulator chains; K unrolled by 2 (K padded to multiple of 8 with zeros).
// Output goes through an LDS transpose buffer so global stores are f-contiguous.
#define KP_MAX 80       // max padded K (3*maxw rounded up to 8), maxw<=26
#define WT_LD 80        // Wt row stride: k and k+2 rows hit disjoint bank windows
#define AG_LD 82        // Ag row stride: even (8B-aligned b64), conflict-light

__global__ __launch_bounds__(128) void mel_gemm_kernel(
    const float* __restrict__ W1, const int* __restrict__ band,
    const float* __restrict__ feat, float* __restrict__ out, int maxw) {
  extern __shared__ float smem[];
  const int WK  = 3 * maxw;
  const int KP8 = (WK + 7) & ~7;
  float* Wt0  = smem;                          // [KP8][WT_LD]
  float* Wt1  = Wt0 + KP_MAX * WT_LD;          // [KP8][WT_LD]
  float* Ag0  = Wt1 + KP_MAX * WT_LD;          // [16][AG_LD]
  float* Ag1  = Ag0 + 16 * AG_LD;              // [16][AG_LD]
  float* OutS = Ag1 + 16 * AG_LD;              // [16 m][64 d][16 f]

  const int tid  = threadIdx.x;
  const int lane = tid & 31;
  const int dt   = tid >> 5;          // wave id = d-tile 0..3
  const int l15  = lane & 15;
  const int koff = (lane >> 4) << 1;  // K sub-offset per fragment layout
  const int bt0  = blockIdx.x * 16;   // 500 blocks over 8000 (b,t) rows
  const int d0   = blockIdx.y * 64;   // 2 blocks over 128 d

  for (int fc = 0; fc < 4; ++fc) {          // 4 chunks of 16 filters
    for (int fp = 0; fp < 8; ++fp) {        // filter pairs within chunk
      const int f0 = fc * 16 + fp * 2;
      const int f1 = f0 + 1;
      __syncthreads();
      // ---- stage Wt{0,1}[k][d] from W1[f][d][c][w] (coalesced global reads) ----
      {
        const float* ws0 = W1 + (size_t)(f0 * D1 + d0) * WK;
        const float* ws1 = W1 + (size_t)(f1 * D1 + d0) * WK;
        for (int j = tid; j < 64 * WK; j += 128) {
          int d = j / WK, jk = j - d * WK;
          Wt0[jk * WT_LD + d] = ws0[j];
          Wt1[jk * WT_LD + d] = ws1[j];
        }
        for (int j = tid; j < 64 * (KP8 - WK); j += 128) {  // zero K padding
          int d = j & 63, k = WK + (j >> 6);
          Wt0[k * WT_LD + d] = 0.0f;
          Wt1[k * WT_LD + d] = 0.0f;
        }
      }
      // ---- stage Ag{0,1}[row][c*maxw+w] = feat window (contiguous gather) ----
      {
        const int bf0 = band[f0 * maxw];   // band_idx[f][0] == window start
        const int bf1 = band[f1 * maxw];
        for (int j = tid; j < 16 * WK; j += 128) {
          int row = j / WK, jk = j - row * WK;
          int c = jk / maxw, w = jk - c * maxw;
          const float* frow = feat + ((size_t)(bt0 + row) * 3 + c) * NFREQ;
          int q0 = bf0 + w; q0 = q0 > (NFREQ - 1) ? (NFREQ - 1) : q0;
          int q1 = bf1 + w; q1 = q1 > (NFREQ - 1) ? (NFREQ - 1) : q1;
          Ag0[row * AG_LD + jk] = frow[q0];
          Ag1[row * AG_LD + jk] = frow[q1];
        }
        for (int j = tid; j < 16 * (KP8 - WK); j += 128) {  // zero (avoid NaN*0)
          int row = j & 15, k = WK + (j >> 4);
          Ag0[row * AG_LD + k] = 0.0f;
          Ag1[row * AG_LD + k] = 0.0f;
        }
      }
      __syncthreads();

      // ---- K loop, unrolled x2: 4 WMMAs, 2 independent accumulator chains ----
      v8f acc0 = {0.f, 0.f, 0.f, 0.f, 0.f, 0.f, 0.f, 0.f};
      v8f acc1 = {0.f, 0.f, 0.f, 0.f, 0.f, 0.f, 0.f, 0.f};
      const int arow = l15 * AG_LD + koff;
      const int bcol = dt * 16 + l15;
      for (int kk = 0; kk < KP8; kk += 8) {
        v2f a00 = *(const v2f*)&Ag0[arow + kk];
        v2f a01 = *(const v2f*)&Ag0[arow + kk + 4];
        v2f a10 = *(const v2f*)&Ag1[arow + kk];
        v2f a11 = *(const v2f*)&Ag1[arow + kk + 4];
        v2f b00, b01, b10, b11;
        b00.x = Wt0[(kk + koff) * WT_LD + bcol];
        b00.y = Wt0[(kk + koff + 1) * WT_LD + bcol];
        b01.x = Wt0[(kk + 4 + koff) * WT_LD + bcol];
        b01.y = Wt0[(kk + 4 + koff + 1) * WT_LD + bcol];
        b10.x = Wt1[(kk + koff) * WT_LD + bcol];
        b10.y = Wt1[(kk + koff + 1) * WT_LD + bcol];
        b11.x = Wt1[(kk + 4 + koff) * WT_LD + bcol];
        b11.y = Wt1[(kk + 4 + koff + 1) * WT_LD + bcol];
        acc0 = __builtin_amdgcn_wmma_f32_16x16x4_f32(
            false, a00, false, b00, (short)0, acc0, false, false);
        acc1 = __builtin_amdgcn_wmma_f32_16x16x4_f32(
            false, a10, false, b10, (short)0, acc1, false, false);
        acc0 = __builtin_amdgcn_wmma_f32_16x16x4_f32(
            false, a01, false, b01, (short)0, acc0, false, false);
        acc1 = __builtin_amdgcn_wmma_f32_16x16x4_f32(
            false, a11, false, b11, (short)0, acc1, false, false);
      }

      // ---- C/D layout: VGPR r -> M = r + 8*(lane>=16), N = lane&15 ----
      const int mh = (lane >> 4) << 3;
      const int fi = fp * 2;
#pragma unroll
      for (int r = 0; r < 8; ++r) {
        float* os = &OutS[((r + mh) * 64 + dt * 16 + l15) * 16 + fi];
        os[0] = acc0[r];
        os[1] = acc1[r];
      }
    }
    __syncthreads();
    // ---- flush: f-contiguous float4 stores (4 consecutive lanes = 64B line) ----
    for (int it = 0; it < 32; ++it) {
      int idx = tid + it * 128;            // 16*64*16/4 = 4096 float4
      int f4 = idx & 3, d = (idx >> 2) & 63, m = idx >> 8;
      int bt = bt0 + m, b = bt / Tsz, t = bt - b * Tsz;
      float4 v = *(const float4*)&OutS[(m * 64 + d) * 16 + f4 * 4];
      *(float4*)&out[((size_t)(b * D1 + d0 + d) * Tsz + t) * NFILTERS +
                     fc * 16 + f4 * 4] = v;
    }
  }
}

extern "C" void kernel_launch(void* const* d_in, const int* in_sizes, int n_in,
                              void* d_out, int out_size, void* d_ws, size_t ws_size,
                              hipStream_t stream) {
  const float* x    = (const float*)d_in[0];
  const float* W1   = (const float*)d_in[1];
  const int*   band = (const int*)d_in[2];
  // d_in[3] = inverse, unused by the reference math
  float* out = (float*)d_out;

  const int maxw = in_sizes[2] / NFILTERS;  // band_idx is [64][maxw]
  const int nx   = in_sizes[0];             // 8*2*1000*257
  unsigned* gmax = (unsigned*)d_ws;
  float* feat = (float*)((char*)d_ws + 256);  // 8000*3*257 f32 scratch (~24.7MB)

  // Phase 0/1: init + global max
  hipMemsetAsync(d_ws, 0, 4, stream);
  mel_max_kernel<<<2048, 256, 0, stream>>>(x, nx, gmax);

  // Phase 2: feature map
  const int n2 = Bsz * Tsz * NFREQ;
  mel_feat_kernel<<<(n2 + 255) / 256, 256, 0, stream>>>(x, gmax, feat, n2);

  // Phase 3: WMMA GEMM, 500 (b,t)-tiles x 2 d-halves
  const size_t smem_bytes =
      (size_t)(2 * KP_MAX * WT_LD + 2 * 16 * AG_LD + 16 * 64 * 16) * sizeof(float);
  dim3 grid3(500, 2, 1);
  mel_gemm_kernel<<<grid3, 128, smem_bytes, stream>>>(W1, band, feat, out, maxw);
}